// LargeLossRejection_32658931319301
// MI455X (gfx1250) — compile-verified
//
#include <hip/hip_runtime.h>
#include <hip/hip_bf16.h>
#include <math.h>

typedef __attribute__((ext_vector_type(4))) float f4;
typedef __attribute__((ext_vector_type(2))) float v2f;
typedef __attribute__((ext_vector_type(8))) float v8f;

#define HIST0_BINS 4096
#define HIST1_BINS 4096
#define HIST2_BINS 256
#define LLR_SENTINEL 0xFFFFFFFFu

__device__ __forceinline__ float llr_softplus(float x) {
  // jax.nn.softplus: max(x,0) + log1p(exp(-|x|)), stable
  return fmaxf(x, 0.0f) + log1pf(expf(-fabsf(x)));
}

// ---------------- Pass 0: zero histograms + meta ----------------
__global__ void llr_zero_meta(unsigned* __restrict__ p, int n) {
  int i = blockIdx.x * blockDim.x + threadIdx.x;
  if (i < n) p[i] = 0u;
}

// ---------------- Pass 1: fused loss + ul + level-0 histogram ----------------
// logits/targets: stream NT (don't pollute L2). out: NT store (write-once).
// ul: regular store -> stays resident in 192MB L2 for the next 3 passes.
__global__ void llr_fuse_hist0(const float* __restrict__ logits,
                               const float* __restrict__ targets,
                               float* __restrict__ out,
                               float* __restrict__ ul,
                               unsigned* __restrict__ hist0,
                               size_t n4) {
  __shared__ unsigned lh[HIST0_BINS];
  for (int i = threadIdx.x; i < HIST0_BINS; i += blockDim.x) lh[i] = 0u;
  __syncthreads();

  const f4* lg = (const f4*)logits;
  const f4* tg = (const f4*)targets;
  f4* o4 = (f4*)out;
  f4* u4 = (f4*)ul;
  size_t stride = (size_t)gridDim.x * blockDim.x;
  for (size_t i = (size_t)blockIdx.x * blockDim.x + threadIdx.x; i < n4; i += stride) {
    if (i + stride < n4) {               // deep prefetch of next tile
      __builtin_prefetch(&lg[i + stride], 0, 0);
      __builtin_prefetch(&tg[i + stride], 0, 0);
    }
    f4 x = __builtin_nontemporal_load(&lg[i]);
    f4 t = __builtin_nontemporal_load(&tg[i]);
    f4 L, U;
#pragma unroll
    for (int c = 0; c < 4; ++c) {
      float xx = x[c], tt = t[c];
      bool unk = (tt != tt);                 // isnan
      float tv = unk ? 0.0f : tt;            // unknown -> negative label
      float l = llr_softplus(xx) - xx * tv;  // BCE with logits
      float u = unk ? l : 0.0f;              // ul = stop_grad(loss) * unknown
      L[c] = l;
      U[c] = u;
      unsigned bits = __float_as_uint(u);
      if (bits != 0u) atomicAdd(&lh[bits >> 20], 1u);  // ul>=0: bits>>20 < 2048
    }
    __builtin_nontemporal_store(L, &o4[i]);
    u4[i] = U;  // keep in L2
  }
  __syncthreads();
  for (int i = threadIdx.x; i < HIST0_BINS; i += blockDim.x) {
    unsigned c = lh[i];
    if (c) atomicAdd(&hist0[i], c);
  }
}

// ---------------- Pass 2: level-0 scan (1 wave). WMMA cross-lane reduce. ----------------
__global__ void llr_scan0(const unsigned* __restrict__ hist0,
                          const int* __restrict__ epoch_p,
                          unsigned* __restrict__ meta) {
  __shared__ float chunks[32];
  int lane = threadIdx.x;
  float psum = 0.0f;
  for (int i = 0; i < HIST0_BINS / 32; ++i)
    psum += (float)hist0[lane * (HIST0_BINS / 32) + i];
  chunks[lane] = psum;

  // n_unknown = sum over all lanes, via V_WMMA_F32_16X16X4_F32:
  // A holds one partial per lane (other K-slots zero), B = all ones
  // => D[m][n] = rowsum_A[m]; summing all 8 D regs + lane^16 gives total.
  v2f a; a.x = psum; a.y = 0.0f;
  v2f b; b.x = 1.0f; b.y = 1.0f;
  v8f cacc = {};
  v8f d = __builtin_amdgcn_wmma_f32_16x16x4_f32(false, a, false, b,
                                                (short)0, cacc, false, false);
  float s = d[0] + d[1] + d[2] + d[3] + d[4] + d[5] + d[6] + d[7];
  float tot = s + __shfl_xor(s, 16, 32);
  __syncthreads();

  if (lane == 0) {
    int epoch = *epoch_p;
    float percent = fminf((float)epoch * 0.1f, 1.0f);
    long long k = (long long)rintf(tot * percent);  // jnp.round = RNE
    if (k < 1) k = 1;
    long long cum = 0;
    int c = 31;
    for (; c >= 0; --c) {
      long long nc = cum + (long long)(chunks[c] + 0.5f);
      if (nc >= k) break;
      cum = nc;
    }
    if (c < 0) {  // k > n_unknown (only when n_unknown==0): threshold = 0
      meta[0] = LLR_SENTINEL;
      meta[4] = 0u;
    } else {
      int b0 = c * 128 + 127;
      for (; b0 > c * 128; --b0) {
        long long nc = cum + (long long)hist0[b0];
        if (nc >= k) break;
        cum = nc;
      }
      meta[0] = (unsigned)b0;
      meta[1] = (unsigned)(k - cum);  // 1-based rank within bin (from top)
    }
  }
}

// ---------------- Pass 3: level-1 histogram (bits[19:8] of prefix-matching ul) ----------------
__global__ void llr_hist1(const float* __restrict__ ul,
                          const unsigned* __restrict__ meta,
                          unsigned* __restrict__ hist1,
                          size_t n4) {
  unsigned b0 = meta[0];
  if (b0 == LLR_SENTINEL) return;
  __shared__ unsigned lh[HIST1_BINS];
  for (int i = threadIdx.x; i < HIST1_BINS; i += blockDim.x) lh[i] = 0u;
  __syncthreads();
  const f4* u4 = (const f4*)ul;
  size_t stride = (size_t)gridDim.x * blockDim.x;
  for (size_t i = (size_t)blockIdx.x * blockDim.x + threadIdx.x; i < n4; i += stride) {
    f4 U = u4[i];  // expected L2 hit
#pragma unroll
    for (int c = 0; c < 4; ++c) {
      unsigned bits = __float_as_uint(U[c]);
      if (bits != 0u && (bits >> 20) == b0)
        atomicAdd(&lh[(bits >> 8) & 0xFFFu], 1u);
    }
  }
  __syncthreads();
  for (int i = threadIdx.x; i < HIST1_BINS; i += blockDim.x) {
    unsigned c = lh[i];
    if (c) atomicAdd(&hist1[i], c);
  }
}

__global__ void llr_scan1(const unsigned* __restrict__ hist1,
                          unsigned* __restrict__ meta) {
  __shared__ float chunks[32];
  int lane = threadIdx.x;
  if (meta[0] == LLR_SENTINEL) {
    if (lane == 0) meta[2] = LLR_SENTINEL;
    return;
  }
  float psum = 0.0f;
  for (int i = 0; i < HIST1_BINS / 32; ++i)
    psum += (float)hist1[lane * (HIST1_BINS / 32) + i];
  chunks[lane] = psum;
  __syncthreads();
  if (lane == 0) {
    long long k = (long long)meta[1];
    long long cum = 0;
    int c = 31;
    for (; c >= 0; --c) {
      long long nc = cum + (long long)(chunks[c] + 0.5f);
      if (nc >= k) break;
      cum = nc;
    }
    int b1 = 0;
    if (c >= 0) {
      b1 = c * 128 + 127;
      for (; b1 > c * 128; --b1) {
        long long nc = cum + (long long)hist1[b1];
        if (nc >= k) break;
        cum = nc;
      }
    }
    meta[2] = (unsigned)b1;
    meta[3] = (unsigned)(k - cum);
  }
}

// ---------------- Pass 4: level-2 histogram (bits[7:0]) ----------------
__global__ void llr_hist2(const float* __restrict__ ul,
                          const unsigned* __restrict__ meta,
                          unsigned* __restrict__ hist2,
                          size_t n4) {
  unsigned b0 = meta[0];
  if (b0 == LLR_SENTINEL) return;
  unsigned prefix = (b0 << 12) | meta[2];  // bits[31:8]
  __shared__ unsigned lh[HIST2_BINS];
  for (int i = threadIdx.x; i < HIST2_BINS; i += blockDim.x) lh[i] = 0u;
  __syncthreads();
  const f4* u4 = (const f4*)ul;
  size_t stride = (size_t)gridDim.x * blockDim.x;
  for (size_t i = (size_t)blockIdx.x * blockDim.x + threadIdx.x; i < n4; i += stride) {
    f4 U = u4[i];
#pragma unroll
    for (int c = 0; c < 4; ++c) {
      unsigned bits = __float_as_uint(U[c]);
      if (bits != 0u && (bits >> 8) == prefix)
        atomicAdd(&lh[bits & 0xFFu], 1u);
    }
  }
  __syncthreads();
  for (int i = threadIdx.x; i < HIST2_BINS; i += blockDim.x) {
    unsigned c = lh[i];
    if (c) atomicAdd(&hist2[i], c);
  }
}

__global__ void llr_scan2(const unsigned* __restrict__ hist2,
                          unsigned* __restrict__ meta) {
  __shared__ float chunks[32];
  int lane = threadIdx.x;
  unsigned b0 = meta[0];
  if (b0 == LLR_SENTINEL) {
    if (lane == 0) meta[4] = 0u;
    return;
  }
  float psum = 0.0f;
  for (int i = 0; i < HIST2_BINS / 32; ++i)
    psum += (float)hist2[lane * (HIST2_BINS / 32) + i];
  chunks[lane] = psum;
  __syncthreads();
  if (lane == 0) {
    long long k = (long long)meta[3];
    long long cum = 0;
    int c = 31;
    for (; c >= 0; --c) {
      long long nc = cum + (long long)(chunks[c] + 0.5f);
      if (nc >= k) break;
      cum = nc;
    }
    int b2 = 0;
    if (c >= 0) {
      b2 = c * 8 + 7;
      for (; b2 > c * 8; --b2) {
        long long nc = cum + (long long)hist2[b2];
        if (nc >= k) break;
        cum = nc;
      }
    }
    meta[4] = (b0 << 20) | (meta[2] << 8) | (unsigned)b2;  // exact k-th largest bits
  }
}

// ---------------- Pass 5: apply lambdas (sparse zeroing of out) ----------------
__global__ void llr_apply(const float* __restrict__ ul,
                          float* __restrict__ out,
                          const unsigned* __restrict__ meta,
                          size_t n4) {
  float thr = __uint_as_float(meta[4]);
  const f4* u4 = (const f4*)ul;
  size_t stride = (size_t)gridDim.x * blockDim.x;
  for (size_t i = (size_t)blockIdx.x * blockDim.x + threadIdx.x; i < n4; i += stride) {
    f4 U = __builtin_nontemporal_load(&u4[i]);  // last use: NT
    size_t base = i * 4;
    if (U[0] > thr) out[base + 0] = 0.0f;
    if (U[1] > thr) out[base + 1] = 0.0f;
    if (U[2] > thr) out[base + 2] = 0.0f;
    if (U[3] > thr) out[base + 3] = 0.0f;
  }
}

extern "C" void kernel_launch(void* const* d_in, const int* in_sizes, int n_in,
                              void* d_out, int out_size, void* d_ws, size_t ws_size,
                              hipStream_t stream) {
  const float* logits  = (const float*)d_in[0];
  const float* targets = (const float*)d_in[1];
  const int*   epoch_p = (const int*)d_in[2];
  float* out = (float*)d_out;

  size_t N  = (size_t)in_sizes[0];
  size_t n4 = N / 4;

  // workspace layout: [ul: N floats][hist0: 4096][hist1: 4096][hist2: 256][meta: 8]
  float*    ul    = (float*)d_ws;
  unsigned* hist0 = (unsigned*)((char*)d_ws + N * sizeof(float));
  unsigned* hist1 = hist0 + HIST0_BINS;
  unsigned* hist2 = hist1 + HIST1_BINS;
  unsigned* meta  = hist2 + HIST2_BINS;
  int metaTotal = HIST0_BINS + HIST1_BINS + HIST2_BINS + 8;

  const int blocks = 4096, threads = 256;

  llr_zero_meta<<<(metaTotal + 255) / 256, 256, 0, stream>>>(hist0, metaTotal);
  llr_fuse_hist0<<<blocks, threads, 0, stream>>>(logits, targets, out, ul, hist0, n4);
  llr_scan0<<<1, 32, 0, stream>>>(hist0, epoch_p, meta);
  llr_hist1<<<blocks, threads, 0, stream>>>(ul, meta, hist1, n4);
  llr_scan1<<<1, 32, 0, stream>>>(hist1, meta);
  llr_hist2<<<blocks, threads, 0, stream>>>(ul, meta, hist2, n4);
  llr_scan2<<<1, 32, 0, stream>>>(hist2, meta);
  llr_apply<<<blocks, threads, 0, stream>>>(ul, out, meta, n4);
}